// reweighting_revision_loss_266287973107
// MI455X (gfx1250) — compile-verified
//
#include <hip/hip_runtime.h>
#include <math.h>

typedef __attribute__((ext_vector_type(2))) float v2f;
typedef __attribute__((ext_vector_type(8))) float v8f;

#define C_DIM 1000
#define B_DIM 32768
#define ROWS_PER_WAVE 16
#define NEG_BIG (-3.402823466e38f)

// ---------------------------------------------------------------------------
// Kernel 1: tiled transpose of T and correction into workspace so the WMMA
// B-matrix gather becomes contiguous per-lane row walks (L0/L2 friendly).
// grid.z = 0 -> T, grid.z = 1 -> correction.
// ---------------------------------------------------------------------------
__global__ __launch_bounds__(1024) void rrl_transpose(const float* __restrict__ T,
                                                      const float* __restrict__ Corr,
                                                      float* __restrict__ Tt,
                                                      float* __restrict__ Ct) {
    __shared__ float tile[32][33];
    const float* src = (blockIdx.z == 0) ? T : Corr;
    float* dst = (blockIdx.z == 0) ? Tt : Ct;

    int x = blockIdx.x * 32 + threadIdx.x;
    int y = blockIdx.y * 32 + threadIdx.y;
    if (x < C_DIM && y < C_DIM)
        tile[threadIdx.y][threadIdx.x] = src[(size_t)y * C_DIM + x];
    __syncthreads();
    int xo = blockIdx.y * 32 + threadIdx.x;
    int yo = blockIdx.x * 32 + threadIdx.y;
    if (xo < C_DIM && yo < C_DIM)
        dst[(size_t)yo * C_DIM + xo] = tile[threadIdx.x][threadIdx.y];
}

// ---------------------------------------------------------------------------
// Kernel 2: one wave32 per 16 rows.
//  Phase 1: per-row softmax max + sum(exp) via shuffle reductions.
//  Phase 2: 250 K-tiles of V_WMMA_F32_16X16X4_F32:
//           A[m][k] = exp(out[row_m][k0+k] - m_m)   (unnormalized probs)
//           B[k][n] = T^T[t_n][k0+k]                (gathered target columns)
//           diagonal of D accumulates the needed dot products.
//  Finalize: beta*ce per row into partial[].
// ---------------------------------------------------------------------------
__global__ __launch_bounds__(256) void rrl_main(const float* __restrict__ outp,
                                                const float* __restrict__ Tt,
                                                const float* __restrict__ Ct,
                                                const int* __restrict__ target,
                                                float* __restrict__ partial) {
    const int lane = threadIdx.x & 31;
    const int waveGlobal = (blockIdx.x * blockDim.x + threadIdx.x) >> 5;
    const int rowBase = waveGlobal * ROWS_PER_WAVE;

    // stats saved for the row this lane represents in the WMMA tiles
    float mA = 0.0f, sA = 1.0f;

    for (int r = 0; r < ROWS_PER_WAVE; ++r) {
        const int row = rowBase + r;
        const float* rp = outp + (size_t)row * C_DIM;
        float vals[32];
        float m = NEG_BIG;
#pragma unroll
        for (int j = 0; j < 32; ++j) {
            int c = lane + 32 * j;
            vals[j] = (c < C_DIM) ? rp[c] : NEG_BIG;
            m = fmaxf(m, vals[j]);
        }
#pragma unroll
        for (int off = 16; off >= 1; off >>= 1)
            m = fmaxf(m, __shfl_xor(m, off));
        float s = 0.0f;
#pragma unroll
        for (int j = 0; j < 32; ++j) {
            int c = lane + 32 * j;
            s += (c < C_DIM) ? __expf(vals[j] - m) : 0.0f;
        }
#pragma unroll
        for (int off = 16; off >= 1; off >>= 1)
            s += __shfl_xor(s, off);
        if ((lane & 15) == r) { mA = m; sA = s; }
    }

    // --- WMMA phase ---
    const int myr  = lane & 15;                 // row index (A) / column slot (B)
    const int rowA = rowBase + myr;
    const int tcol = target[rowA];              // this slot's gathered column
    const int koff = (lane < 16) ? 0 : 2;       // A 16x4 f32 layout: low lanes K=0/1, high lanes K=2/3

    const float* aPtr  = outp + (size_t)rowA * C_DIM + koff;
    const float* bTPtr = Tt   + (size_t)tcol * C_DIM + koff;
    const float* bCPtr = Ct   + (size_t)tcol * C_DIM + koff;

    v8f accT = {0.f, 0.f, 0.f, 0.f, 0.f, 0.f, 0.f, 0.f};
    v8f accC = {0.f, 0.f, 0.f, 0.f, 0.f, 0.f, 0.f, 0.f};

    for (int k0 = 0; k0 < C_DIM; k0 += 4) {
        float2 av = *reinterpret_cast<const float2*>(aPtr + k0);
        float2 tv = *reinterpret_cast<const float2*>(bTPtr + k0);
        float2 cv = *reinterpret_cast<const float2*>(bCPtr + k0);
        v2f a, bT, bC;
        a.x = __expf(av.x - mA);
        a.y = __expf(av.y - mA);
        bT.x = tv.x; bT.y = tv.y;
        bC.x = cv.x; bC.y = cv.y;
        // D = A x B + C  (8 args: neg_a, A, neg_b, B, c_mod, C, reuse_a, reuse_b)
        accT = __builtin_amdgcn_wmma_f32_16x16x4_f32(false, a, false, bT, (short)0, accT, false, false);
        accC = __builtin_amdgcn_wmma_f32_16x16x4_f32(false, a, false, bC, (short)0, accC, false, false);
    }

    // Diagonal extraction (C/D layout: VGPR r -> M=r (lanes 0-15, N=lane),
    // M=r+8 (lanes 16-31, N=lane-16); diag m<8 at (vgpr m, lane m),
    // diag m>=8 at (vgpr m-8, lane m+16)).
    float myDT = 0.0f, myDC = 0.0f;
#pragma unroll
    for (int m = 0; m < 8; ++m) {
        float vT  = __shfl(accT[m], m);
        float vC  = __shfl(accC[m], m);
        float vT2 = __shfl(accT[m], m + 24);
        float vC2 = __shfl(accC[m], m + 24);
        if (lane == m)     { myDT = vT;  myDC = vC;  }
        if (lane == m + 8) { myDT = vT2; myDC = vC2; }
    }

    if (lane < 16) {
        const int row = rowBase + lane;
        const float logit_t = outp[(size_t)row * C_DIM + tcol];
        const float et    = __expf(logit_t - mA);           // unnormalized prob at label
        const float ce    = -(logit_t - mA - __logf(sA));   // -log softmax at label
        const float scale = (float)(row + 1);
        // beta = (et/S) / ((dT + scale*dC)/S) = et / (dT + scale*dC)
        const float beta  = et / (myDT + scale * myDC);
        partial[row] = beta * ce;
    }
}

// ---------------------------------------------------------------------------
// Kernel 3: deterministic fixed-order reduction -> mean.
// ---------------------------------------------------------------------------
__global__ __launch_bounds__(256) void rrl_reduce(const float* __restrict__ partial,
                                                  float* __restrict__ outv) {
    __shared__ float sh[256];
    float s = 0.0f;
    for (int i = threadIdx.x; i < B_DIM; i += 256)
        s += partial[i];
    sh[threadIdx.x] = s;
    __syncthreads();
    for (int stride = 128; stride >= 1; stride >>= 1) {
        if ((int)threadIdx.x < stride)
            sh[threadIdx.x] += sh[threadIdx.x + stride];
        __syncthreads();
    }
    if (threadIdx.x == 0)
        outv[0] = sh[0] * (1.0f / (float)B_DIM);
}

// ---------------------------------------------------------------------------
extern "C" void kernel_launch(void* const* d_in, const int* in_sizes, int n_in,
                              void* d_out, int out_size, void* d_ws, size_t ws_size,
                              hipStream_t stream) {
    const float* outp = (const float*)d_in[0];
    const float* T    = (const float*)d_in[1];
    const float* corr = (const float*)d_in[2];
    const int*   tgt  = (const int*)d_in[3];

    float* wsf     = (float*)d_ws;
    float* Tt      = wsf;
    float* Ct      = wsf + (size_t)C_DIM * C_DIM;
    float* partial = wsf + 2 * (size_t)C_DIM * C_DIM;

    dim3 tb(32, 32, 1);
    dim3 tg((C_DIM + 31) / 32, (C_DIM + 31) / 32, 2);
    rrl_transpose<<<tg, tb, 0, stream>>>(T, corr, Tt, Ct);

    const int waves = B_DIM / ROWS_PER_WAVE;        // 2048
    const int blocks = waves / 8;                   // 256 blocks of 256 threads (8 waves)
    rrl_main<<<dim3(blocks), dim3(256), 0, stream>>>(outp, Tt, Ct, tgt, partial);

    rrl_reduce<<<dim3(1), dim3(256), 0, stream>>>(partial, (float*)d_out);
}